// VoxelPooling_35347580846886
// MI455X (gfx1250) — compile-verified
//
#include <hip/hip_runtime.h>

typedef float v2f __attribute__((ext_vector_type(2)));
typedef float v8f __attribute__((ext_vector_type(8)));

#define NPTS   1000000
#define CIN    32
#define COUT   64
#define NB     4
#define GRIDSZ 0.05f
#define EPSV   1e-5f
#define MAXKEY (1 << 17)

#define WMMA_F32(a, b, c) \
  __builtin_amdgcn_wmma_f32_16x16x4_f32(false, (a), false, (b), (short)0, (c), false, false)

// ---------------- generic fills ----------------
__global__ void ffill_kernel(float* p, long n, float v) {
  long i = (long)blockIdx.x * blockDim.x + threadIdx.x;
  long stride = (long)gridDim.x * blockDim.x;
  for (; i < n; i += stride) p[i] = v;
}
__global__ void ifill_kernel(int* p, long n, int v) {
  long i = (long)blockIdx.x * blockDim.x + threadIdx.x;
  long stride = (long)gridDim.x * blockDim.x;
  for (; i < n; i += stride) p[i] = v;
}

__device__ __forceinline__ int point_batch(long i, const long long* __restrict__ off) {
  int b = 0;
#pragma unroll
  for (int j = 0; j < NB - 1; ++j) b += (i >= off[j]) ? 1 : 0;
  return b;
}

// ---------------- pass 1: WMMA GEMM + BN statistics ----------------
__global__ void __launch_bounds__(256) gemm_stats_kernel(
    const float* __restrict__ feat, const float* __restrict__ W,
    float* __restrict__ gsum, float* __restrict__ gsq) {
  __shared__ float sW[CIN * COUT];
  __shared__ float lsum[COUT];
  __shared__ float lsq[COUT];
  int tid = threadIdx.x;
  for (int i = tid; i < CIN * COUT; i += 256) sW[i] = W[i];
  if (tid < COUT) { lsum[tid] = 0.f; lsq[tid] = 0.f; }
  __syncthreads();

  int wave = tid >> 5;
  int lane = tid & 31;
  long m0 = ((long)blockIdx.x * 8 + wave) * 16;
  if (m0 + 16 <= NPTS) {               // wave-uniform guard: EXEC all-ones inside
    int row   = lane & 15;
    int col   = lane & 15;
    int khalf = (lane >> 4) << 1;      // lanes 0-15 -> K 0,1 ; lanes 16-31 -> K 2,3
    const float* arow = feat + (m0 + row) * CIN;
    v8f acc[4] = {};
#pragma unroll
    for (int k = 0; k < CIN; k += 4) {
      v2f a;
      a.x = arow[k + khalf];
      a.y = arow[k + khalf + 1];
      const float* b0 = sW + (k + khalf) * COUT + col;
      const float* b1 = b0 + COUT;
#pragma unroll
      for (int t = 0; t < 4; ++t) {
        v2f b;
        b.x = b0[t * 16];
        b.y = b1[t * 16];
        acc[t] = WMMA_F32(a, b, acc[t]);
      }
    }
#pragma unroll
    for (int t = 0; t < 4; ++t) {
      int colg = t * 16 + col;
      float s = 0.f, q = 0.f;
#pragma unroll
      for (int r = 0; r < 8; ++r) { float v = acc[t][r]; s += v; q += v * v; }
      atomicAdd(&lsum[colg], s);
      atomicAdd(&lsq[colg], q);
    }
  }
  __syncthreads();
  if (tid < COUT) {
    atomicAdd(&gsum[tid], lsum[tid]);
    atomicAdd(&gsq[tid], lsq[tid]);
  }
}

__global__ void finalize_stats_kernel(const float* __restrict__ gsum,
                                      const float* __restrict__ gsq,
                                      const float* __restrict__ gamma,
                                      const float* __restrict__ beta,
                                      float* __restrict__ ascale,
                                      float* __restrict__ bbias) {
  int c = threadIdx.x;
  if (c < COUT) {
    float mean = gsum[c] / (float)NPTS;
    float var  = gsq[c] / (float)NPTS - mean * mean;
    float sc   = gamma[c] * rsqrtf(var + EPSV);
    ascale[c]  = sc;
    bbias[c]   = beta[c] - mean * sc;
  }
}

// ---------------- voxelization ----------------
__global__ void batch_min_kernel(const float* __restrict__ coord,
                                 const long long* __restrict__ off,
                                 unsigned* __restrict__ bmin) {
  long i = (long)blockIdx.x * blockDim.x + threadIdx.x;
  if (i >= NPTS) return;
  int b = point_batch(i, off);
#pragma unroll
  for (int j = 0; j < 3; ++j)
    atomicMin(&bmin[b * 3 + j], __float_as_uint(coord[i * 3 + j]));  // coords >= 0
}

__global__ void vox_dims_kernel(const float* __restrict__ coord,
                                const long long* __restrict__ off,
                                const unsigned* __restrict__ bmin,
                                int* __restrict__ dims) {
  long i = (long)blockIdx.x * blockDim.x + threadIdx.x;
  if (i >= NPTS) return;
  int b = point_batch(i, off);
#pragma unroll
  for (int j = 0; j < 3; ++j) {
    float pos = coord[i * 3 + j] - __uint_as_float(bmin[b * 3 + j]);
    int v = (int)floorf(pos / GRIDSZ);
    if (v < 0) v = 0;
    atomicMax(&dims[j], v);
  }
}

__global__ void keys_hist_kernel(const float* __restrict__ coord,
                                 const long long* __restrict__ off,
                                 const unsigned* __restrict__ bmin,
                                 const int* __restrict__ dims,
                                 int* __restrict__ keyarr,
                                 int* __restrict__ bincount) {
  long i = (long)blockIdx.x * blockDim.x + threadIdx.x;
  if (i >= NPTS) return;
  int b = point_batch(i, off);
  int d0 = dims[0] + 1, d1 = dims[1] + 1, d2 = dims[2] + 1;
  int vx[3];
#pragma unroll
  for (int j = 0; j < 3; ++j) {
    float pos = coord[i * 3 + j] - __uint_as_float(bmin[b * 3 + j]);
    int v = (int)floorf(pos / GRIDSZ);
    if (v < 0) v = 0;
    if (v > dims[j]) v = dims[j];
    vx[j] = v;
  }
  int k = ((b * d0 + vx[0]) * d1 + vx[1]) * d2 + vx[2];
  if (k < 0) k = 0;
  if (k >= MAXKEY) k = MAXKEY - 1;
  keyarr[i] = k;
  atomicAdd(&bincount[k], 1);
}

// exclusive prefix sum of occupancy flags -> sorted-unique rank per key
__global__ void __launch_bounds__(1024) scan_kernel(const int* __restrict__ bincount,
                                                    int* __restrict__ rank,
                                                    int* __restrict__ nc) {
  __shared__ int buf[1024];
  __shared__ int running;
  int tid = threadIdx.x;
  if (tid == 0) running = 0;
  __syncthreads();
  for (int base = 0; base < MAXKEY; base += 1024) {
    int flag = bincount[base + tid] > 0 ? 1 : 0;
    buf[tid] = flag;
    __syncthreads();
    for (int o = 1; o < 1024; o <<= 1) {
      int v = (tid >= o) ? buf[tid - o] : 0;
      __syncthreads();
      buf[tid] += v;
      __syncthreads();
    }
    rank[base + tid] = running + buf[tid] - flag;
    __syncthreads();
    if (tid == 0) running += buf[1023];
    __syncthreads();
  }
  if (tid == 0) *nc = running;
}

__global__ void scatter_coord_kernel(const float* __restrict__ coord,
                                     const int* __restrict__ keyarr,
                                     const int* __restrict__ rank,
                                     int* __restrict__ invi,
                                     float* __restrict__ inv_out,
                                     float* __restrict__ coord_out) {
  long i = (long)blockIdx.x * blockDim.x + threadIdx.x;
  if (i >= NPTS) return;
  int c = rank[keyarr[i]];
  invi[i] = c;
  inv_out[i] = (float)c;
#pragma unroll
  for (int j = 0; j < 3; ++j)
    atomicAdd(&coord_out[(long)c * 3 + j], coord[i * 3 + j]);
}

__global__ void cluster_info_kernel(const int* __restrict__ bincount,
                                    const int* __restrict__ rank,
                                    const int* __restrict__ dims,
                                    float* __restrict__ counts_out,
                                    float* __restrict__ batch_out) {
  long k = (long)blockIdx.x * blockDim.x + threadIdx.x;
  if (k >= MAXKEY) return;
  int cnt = bincount[k];
  if (cnt > 0) {
    int c = rank[k];
    counts_out[c] = (float)cnt;
    int d0 = dims[0] + 1, d1 = dims[1] + 1, d2 = dims[2] + 1;
    batch_out[c] = (float)((int)(k / (d0 * d1 * d2)));
  }
}

__global__ void divide_kernel(float* __restrict__ coord_out,
                              const float* __restrict__ counts_out) {
  long c = (long)blockIdx.x * blockDim.x + threadIdx.x;
  if (c >= NPTS) return;
  float cnt = counts_out[c];
  if (cnt > 0.f) {
    float s = 1.f / cnt;
#pragma unroll
    for (int j = 0; j < 3; ++j) coord_out[c * 3 + j] *= s;
  }
}

// ---------------- pass 2: WMMA GEMM + BN apply + ReLU + segment-max scatter ----------------
__global__ void __launch_bounds__(256) gemm_scatter_kernel(
    const float* __restrict__ feat, const float* __restrict__ W,
    const float* __restrict__ ascale, const float* __restrict__ bbias,
    const int* __restrict__ invi, float* __restrict__ feat_out) {
  __shared__ float sW[CIN * COUT];
  __shared__ float sA[COUT];
  __shared__ float sB[COUT];
  int tid = threadIdx.x;
  for (int i = tid; i < CIN * COUT; i += 256) sW[i] = W[i];
  if (tid < COUT) { sA[tid] = ascale[tid]; sB[tid] = bbias[tid]; }
  __syncthreads();

  int wave = tid >> 5;
  int lane = tid & 31;
  long m0 = ((long)blockIdx.x * 8 + wave) * 16;
  if (m0 + 16 > NPTS) return;          // after the only barrier; wave-uniform

  int row   = lane & 15;
  int col   = lane & 15;
  int khalf = (lane >> 4) << 1;
  const float* arow = feat + (m0 + row) * CIN;
  v8f acc[4] = {};
#pragma unroll
  for (int k = 0; k < CIN; k += 4) {
    v2f a;
    a.x = arow[k + khalf];
    a.y = arow[k + khalf + 1];
    const float* b0 = sW + (k + khalf) * COUT + col;
    const float* b1 = b0 + COUT;
#pragma unroll
    for (int t = 0; t < 4; ++t) {
      v2f b;
      b.x = b0[t * 16];
      b.y = b1[t * 16];
      acc[t] = WMMA_F32(a, b, acc[t]);
    }
  }
  long mbase = m0 + ((lane >> 4) << 3);   // D rows r (lanes 0-15) / r+8 (lanes 16-31)
#pragma unroll
  for (int t = 0; t < 4; ++t) {
    int colg = t * 16 + col;
    float sc = sA[colg], bi = sB[colg];
#pragma unroll
    for (int r = 0; r < 8; ++r) {
      float v = sc * acc[t][r] + bi;
      v = v > 0.f ? v : 0.f;               // ReLU -> non-negative, uint max == float max
      long c = invi[mbase + r];
      atomicMax((unsigned int*)&feat_out[c * COUT + colg], __float_as_uint(v));
    }
  }
}

// ---------------- launcher ----------------
extern "C" void kernel_launch(void* const* d_in, const int* in_sizes, int n_in,
                              void* d_out, int out_size, void* d_ws, size_t ws_size,
                              hipStream_t stream) {
  const float*     coord  = (const float*)d_in[0];
  const float*     feat   = (const float*)d_in[1];
  const long long* offset = (const long long*)d_in[2];   // jnp int64
  const float*     W      = (const float*)d_in[3];
  const float*     gamma  = (const float*)d_in[4];
  const float*     beta   = (const float*)d_in[5];

  float* out        = (float*)d_out;
  float* coord_out  = out;                         // N*3
  float* feat_out   = out + (size_t)3 * NPTS;      // N*64
  float* batch_out  = out + (size_t)67 * NPTS;     // N
  float* inv_out    = out + (size_t)68 * NPTS;     // N
  float* counts_out = out + (size_t)69 * NPTS;     // N

  float*    gsum     = (float*)d_ws;               // 64
  float*    gsq      = gsum + 64;                  // 64
  float*    ascale   = gsum + 128;                 // 64
  float*    bbias    = gsum + 192;                 // 64
  unsigned* bmin     = (unsigned*)(gsum + 256);    // 12
  int*      dims     = (int*)(bmin + 12);          // 3
  int*      ncp      = dims + 3;                   // 1
  int*      keyarr   = (int*)((char*)d_ws + 2048); // N
  int*      bincount = keyarr + NPTS;              // MAXKEY
  int*      rankarr  = bincount + MAXKEY;          // MAXKEY
  int*      invi     = rankarr + MAXKEY;           // N

  const int nblk       = (NPTS + 255) / 256;
  const int gemmBlocks = (NPTS / 16 + 7) / 8;

  // init (harness poisons buffers; re-init every call, deterministically)
  ffill_kernel<<<1, 256, 0, stream>>>(gsum, 128, 0.f);
  ifill_kernel<<<1, 64, 0, stream>>>(dims, 4, 0);                       // dims + nc
  ifill_kernel<<<1, 64, 0, stream>>>((int*)bmin, 12, 0x7F800000);       // +inf bits
  ifill_kernel<<<(MAXKEY + 255) / 256, 256, 0, stream>>>(bincount, MAXKEY, 0);
  ffill_kernel<<<2048, 256, 0, stream>>>(coord_out, (long)3 * NPTS, 0.f);
  ffill_kernel<<<8192, 256, 0, stream>>>(feat_out, (long)COUT * NPTS, 0.f);
  ffill_kernel<<<2048, 256, 0, stream>>>(counts_out, NPTS, 0.f);
  ffill_kernel<<<2048, 256, 0, stream>>>(batch_out, NPTS, (float)NB);   // invalid -> B

  // GEMM pass 1: statistics
  gemm_stats_kernel<<<gemmBlocks, 256, 0, stream>>>(feat, W, gsum, gsq);
  finalize_stats_kernel<<<1, 64, 0, stream>>>(gsum, gsq, gamma, beta, ascale, bbias);

  // voxel pipeline
  batch_min_kernel<<<nblk, 256, 0, stream>>>(coord, offset, bmin);
  vox_dims_kernel<<<nblk, 256, 0, stream>>>(coord, offset, bmin, dims);
  keys_hist_kernel<<<nblk, 256, 0, stream>>>(coord, offset, bmin, dims, keyarr, bincount);
  scan_kernel<<<1, 1024, 0, stream>>>(bincount, rankarr, ncp);
  scatter_coord_kernel<<<nblk, 256, 0, stream>>>(coord, keyarr, rankarr, invi, inv_out, coord_out);
  cluster_info_kernel<<<(MAXKEY + 255) / 256, 256, 0, stream>>>(bincount, rankarr, dims,
                                                                counts_out, batch_out);
  divide_kernel<<<nblk, 256, 0, stream>>>(coord_out, counts_out);

  // GEMM pass 2: BN apply + ReLU + segment-max scatter
  gemm_scatter_kernel<<<gemmBlocks, 256, 0, stream>>>(feat, W, ascale, bbias, invi, feat_out);
}